// DVGRL_17755394802209
// MI455X (gfx1250) — compile-verified
//
#include <hip/hip_runtime.h>
#include <hip/hip_bf16.h>

typedef __attribute__((ext_vector_type(2))) float v2f;
typedef __attribute__((ext_vector_type(8))) float v8f;

#define U_DIM 30000
#define I_DIM 20000
#define E_DIM 64
#define B_DIM 4096
#define F_DIM 128   // 2*E

// ---------------------------------------------------------------------------
// K0: transpose W (128 x C) -> Wt (C x 128) so SpMM reads contiguous rows
// ---------------------------------------------------------------------------
__global__ void k_transpose128(const float* __restrict__ W, float* __restrict__ Wt, int C) {
    int t = blockIdx.x * blockDim.x + threadIdx.x;
    int total = C * F_DIM;
    if (t >= total) return;
    int c = t >> 7;        // column of W
    int f = t & 127;       // row of W
    Wt[t] = W[(size_t)f * C + c];   // Wt[c*128 + f] = W[f, c]
}

// ---------------------------------------------------------------------------
// K1: H[u,f] = bq[f];  Hs[u,f] = bs[f]
// ---------------------------------------------------------------------------
__global__ void k_init_bias(float* __restrict__ H, float* __restrict__ Hs,
                            const float* __restrict__ bq, const float* __restrict__ bs) {
    int t = blockIdx.x * blockDim.x + threadIdx.x;
    if (t >= U_DIM * F_DIM) return;
    int f = t & 127;
    H[t]  = bq[f];
    Hs[t] = bs[f];
}

// ---------------------------------------------------------------------------
// K2: scatter SpMM: H[row, :] += val * Wt[col, :]   (128 lanes = one edge row)
// ---------------------------------------------------------------------------
#define EDGES_PER_BLOCK 8
__global__ void k_spmm_scatter(const int* __restrict__ rows, const int* __restrict__ cols,
                               const float* __restrict__ vals, const float* __restrict__ Wt,
                               float* __restrict__ H, int nnz) {
    int f = threadIdx.x;                       // 0..127
    int e0 = blockIdx.x * EDGES_PER_BLOCK;
    #pragma unroll
    for (int i = 0; i < EDGES_PER_BLOCK; ++i) {
        int e = e0 + i;
        if (e >= nnz) return;
        int   r = rows[e];
        int   c = cols[e];
        float v = vals[e];
        atomicAdd(&H[(size_t)r * F_DIM + f], v * Wt[(size_t)c * F_DIM + f]);
    }
}

// ---------------------------------------------------------------------------
// K3: split H/Hs into mu / logvar / s_mu / s_logvar outputs
// ---------------------------------------------------------------------------
__global__ void k_split(const float* __restrict__ H, const float* __restrict__ Hs,
                        float* __restrict__ mu, float* __restrict__ logvar,
                        float* __restrict__ smu, float* __restrict__ slogvar) {
    int t = blockIdx.x * blockDim.x + threadIdx.x;
    if (t >= U_DIM * E_DIM) return;
    int u = t >> 6;
    int f = t & 63;
    size_t base = (size_t)u * F_DIM;
    mu[t]      = H[base + f];
    logvar[t]  = H[base + 64 + f];
    smu[t]     = Hs[base + f];
    slogvar[t] = Hs[base + 64 + f];
}

// ---------------------------------------------------------------------------
// K4: gather + reparam:  u_z = mu[inp] + exp(0.5*logvar[inp]); same for s_z
// ---------------------------------------------------------------------------
__global__ void k_gather_z(const int* __restrict__ inputs,
                           const float* __restrict__ H, const float* __restrict__ Hs,
                           float* __restrict__ uz, float* __restrict__ sz) {
    int t = blockIdx.x * blockDim.x + threadIdx.x;
    if (t >= B_DIM * E_DIM) return;
    int b = t >> 6;
    int f = t & 63;
    size_t base = (size_t)inputs[b] * F_DIM;
    uz[t] = H[base + f]  + expf(0.5f * H[base + 64 + f]);
    sz[t] = Hs[base + f] + expf(0.5f * Hs[base + 64 + f]);
}

// ---------------------------------------------------------------------------
// K5: attention gate + mix:  score = tanh(all_z @ aW1.T + ab1) @ aW2.T
//     z = score*u_z + (1-score)*s_z       (one block of 64 threads per batch b)
// ---------------------------------------------------------------------------
__global__ void __launch_bounds__(64) k_attention(const float* __restrict__ uz,
                                                  const float* __restrict__ sz,
                                                  const float* __restrict__ aW1,
                                                  const float* __restrict__ ab1,
                                                  const float* __restrict__ aW2,
                                                  float* __restrict__ z) {
    __shared__ float allz[F_DIM];
    __shared__ float red[E_DIM];
    int b = blockIdx.x;
    int j = threadIdx.x;      // 0..63
    allz[j]      = uz[(size_t)b * E_DIM + j];
    allz[64 + j] = sz[(size_t)b * E_DIM + j];
    __syncthreads();
    float acc = ab1[j];
    #pragma unroll 8
    for (int k = 0; k < F_DIM; ++k)
        acc += aW1[j * F_DIM + k] * allz[k];
    red[j] = tanhf(acc) * aW2[j];
    __syncthreads();
    for (int s = 32; s > 0; s >>= 1) {
        if (j < s) red[j] += red[j + s];
        __syncthreads();
    }
    float score = red[0];
    z[(size_t)b * E_DIM + j] = score * allz[j] + (1.0f - score) * allz[64 + j];
}

// ---------------------------------------------------------------------------
// K6: WMMA GEMM:  C(M x N) = A(M x 64) @ Bt(N x 64)^T + bias(N)
//     one wave per 16x16 tile, K=64 via 16x V_WMMA_F32_16X16X4_F32
//     A layout : lane L -> m = L%16, k = 4*ks + 2*(L/16) + v   (v in {0,1})
//     B layout : lane L -> n = L%16, same k mapping (Bt row-major = B^T)
//     C layout : lane L -> n = L%16, m = v + 8*(L/16), v in 0..7
// ---------------------------------------------------------------------------
__global__ void __launch_bounds__(256)
k_gemm_nt64_wmma(const float* __restrict__ A, const float* __restrict__ Bt,
                 const float* __restrict__ bias, float* __restrict__ C,
                 int M, int N) {
    const int lane   = threadIdx.x & 31;
    const int wave   = threadIdx.x >> 5;
    const int ntiles = N >> 4;
    const int tile   = blockIdx.x * 8 + wave;
    const int mt     = tile / ntiles;
    const int nt     = tile - mt * ntiles;
    if (mt >= (M >> 4)) return;

    const int half = lane >> 4;    // 0 or 1 (selects k pair)
    const int lid  = lane & 15;    // m for A, n for B/C

    const float* Arow = A  + ((size_t)(mt * 16 + lid)) * 64 + 2 * half;
    const float* Brow = Bt + ((size_t)(nt * 16 + lid)) * 64 + 2 * half;

    v8f c = {};
    #pragma unroll
    for (int ks = 0; ks < 16; ++ks) {
        v2f a = *(const v2f*)(Arow + ks * 4);
        v2f b = *(const v2f*)(Brow + ks * 4);
        c = __builtin_amdgcn_wmma_f32_16x16x4_f32(
                /*neg_a=*/false, a, /*neg_b=*/false, b,
                /*c_mod=*/(short)0, c, /*reuse_a=*/false, /*reuse_b=*/false);
    }

    const float bv = bias[nt * 16 + lid];
    float* Cp = C + (size_t)(nt * 16 + lid);
    #pragma unroll
    for (int v = 0; v < 8; ++v) {
        int m = mt * 16 + v + 8 * half;
        Cp[(size_t)m * N] = c[v] + bv;
    }
}

// ---------------------------------------------------------------------------
extern "C" void kernel_launch(void* const* d_in, const int* in_sizes, int n_in,
                              void* d_out, int out_size, void* d_ws, size_t ws_size,
                              hipStream_t stream) {
    const int*   inputs    = (const int*)  d_in[0];
    const int*   graph_row = (const int*)  d_in[1];
    const int*   graph_col = (const int*)  d_in[2];
    const float* graph_val = (const float*)d_in[3];
    const int*   s_row     = (const int*)  d_in[4];
    const int*   s_col     = (const int*)  d_in[5];
    const float* s_val     = (const float*)d_in[6];
    const float* Wq        = (const float*)d_in[7];    // (128, I)
    const float* bq        = (const float*)d_in[8];
    const float* Ws        = (const float*)d_in[9];    // (128, U)
    const float* bs        = (const float*)d_in[10];
    const float* aW1       = (const float*)d_in[11];   // (64, 128)
    const float* ab1       = (const float*)d_in[12];
    const float* aW2       = (const float*)d_in[13];   // (1, 64)
    const float* Wp        = (const float*)d_in[14];   // (I, 64)  == recon_A B^T
    const float* bp        = (const float*)d_in[15];
    const float* Wsp       = (const float*)d_in[16];   // (U, 64)  == recon_S B^T
    const float* bsp       = (const float*)d_in[17];

    const int nnz  = in_sizes[1];
    const int snnz = in_sizes[4];

    // ---- output layout (floats, concatenated in reference return order) ----
    float* out      = (float*)d_out;
    float* reconA   = out;                                  // B*I
    float* reconS   = reconA  + (size_t)B_DIM * I_DIM;      // B*U
    float* o_mu     = reconS  + (size_t)B_DIM * U_DIM;      // U*E
    float* o_logvar = o_mu    + (size_t)U_DIM * E_DIM;
    float* o_smu    = o_logvar+ (size_t)U_DIM * E_DIM;
    float* o_slogv  = o_smu   + (size_t)U_DIM * E_DIM;
    float* o_uz     = o_slogv + (size_t)U_DIM * E_DIM;      // B*E
    float* o_sz     = o_uz    + (size_t)B_DIM * E_DIM;      // B*E

    // ---- workspace layout ----
    float* ws   = (float*)d_ws;
    float* WqT  = ws;                                  // I*128
    float* WsT  = WqT  + (size_t)I_DIM * F_DIM;        // U*128
    float* Hbuf = WsT  + (size_t)U_DIM * F_DIM;        // U*128
    float* Hsb  = Hbuf + (size_t)U_DIM * F_DIM;        // U*128
    float* zbuf = Hsb  + (size_t)U_DIM * F_DIM;        // B*64

    // 0) transposed weights for coalesced SpMM reads
    {
        int n = I_DIM * F_DIM;
        k_transpose128<<<(n + 255) / 256, 256, 0, stream>>>(Wq, WqT, I_DIM);
        n = U_DIM * F_DIM;
        k_transpose128<<<(n + 255) / 256, 256, 0, stream>>>(Ws, WsT, U_DIM);
    }
    // 1) H = bq, Hs = bs
    {
        int n = U_DIM * F_DIM;
        k_init_bias<<<(n + 255) / 256, 256, 0, stream>>>(Hbuf, Hsb, bq, bs);
    }
    // 2) scatter SpMMs
    k_spmm_scatter<<<(nnz  + EDGES_PER_BLOCK - 1) / EDGES_PER_BLOCK, F_DIM, 0, stream>>>(
        graph_row, graph_col, graph_val, WqT, Hbuf, nnz);
    k_spmm_scatter<<<(snnz + EDGES_PER_BLOCK - 1) / EDGES_PER_BLOCK, F_DIM, 0, stream>>>(
        s_row, s_col, s_val, WsT, Hsb, snnz);
    // 3) split into mu/logvar/s_mu/s_logvar outputs
    {
        int n = U_DIM * E_DIM;
        k_split<<<(n + 255) / 256, 256, 0, stream>>>(Hbuf, Hsb, o_mu, o_logvar, o_smu, o_slogv);
    }
    // 4) gather + reparameterize
    {
        int n = B_DIM * E_DIM;
        k_gather_z<<<(n + 255) / 256, 256, 0, stream>>>(inputs, Hbuf, Hsb, o_uz, o_sz);
    }
    // 5) attention gate + mix -> z
    k_attention<<<B_DIM, 64, 0, stream>>>(o_uz, o_sz, aW1, ab1, aW2, zbuf);
    // 6) recon_A = z @ Wp.T + bp   (M=4096, N=20000, K=64) — WMMA
    {
        int tiles = (B_DIM / 16) * (I_DIM / 16);   // 320000, divisible by 8
        k_gemm_nt64_wmma<<<tiles / 8, 256, 0, stream>>>(zbuf, Wp, bp, reconA, B_DIM, I_DIM);
    }
    // 7) recon_S = s_z @ Wsp.T + bsp  (M=4096, N=30000, K=64) — WMMA
    {
        int tiles = (B_DIM / 16) * (U_DIM / 16);   // 480000, divisible by 8
        k_gemm_nt64_wmma<<<tiles / 8, 256, 0, stream>>>(o_sz, Wsp, bsp, reconS, B_DIM, U_DIM);
    }
}